// DGLRGCNHistory_39522289058162
// MI455X (gfx1250) — compile-verified
//
#include <hip/hip_runtime.h>
#include <hip/hip_bf16.h>

typedef __attribute__((ext_vector_type(2))) float v2f;
typedef __attribute__((ext_vector_type(8))) float v8f;

// ---------------------------------------------------------------------------
// Kernel 1: per-relation node transform via fp32 WMMA (16x16x4).
// One wave handles one (16-node tile, relation) job, producing a 16x64 tile.
// r in [0,7]  -> h_all[node][r][h]   (node stride 512 floats)
// r == 8      -> agg[node][h] = x @ loop_w + bias   (segment-sum init)
// ---------------------------------------------------------------------------
__global__ __launch_bounds__(256) void rgcn_mm(
    const float* __restrict__ x,       // [N, 64]
    const float* __restrict__ W,       // [8, 64, 64]
    const float* __restrict__ loop_w,  // [64, 64]
    const float* __restrict__ bias,    // [64]
    float* __restrict__ h_all,         // [N, 8, 64]
    float* __restrict__ agg,           // [N, 64]
    int n_tiles, int n_nodes)
{
    const int wave = blockIdx.x * (blockDim.x >> 5) + (threadIdx.x >> 5);
    const int jobs = n_tiles * 9;
    if (wave >= jobs) return;                 // uniform per wave: EXEC stays all-1s

    const int r    = wave % 9;
    const int tile = wave / 9;
    const int m0   = tile * 16;

    const int lane = threadIdx.x & 31;
    const int half = lane >> 4;               // 0: lanes 0-15, 1: lanes 16-31
    const int l15  = lane & 15;

    // A fragment source row (clamped for safety; N%16==0 in practice)
    int arow = m0 + l15;
    if (arow >= n_nodes) arow = n_nodes - 1;
    const float* xrow = x + (size_t)arow * 64 + 2 * half;  // a = {x[row][k+2h], x[row][k+2h+1]}

    const float* Wr = (r < 8) ? (W + (size_t)r * 4096) : loop_w;  // [K=64][N=64]

    v8f c0 = {}, c1 = {}, c2 = {}, c3 = {};

    for (int k = 0; k < 64; k += 4) {
        // A 16x4 fragment: VGPR0 = K=k+2*half, VGPR1 = K=k+2*half+1 (ISA 7.12.2)
        v2f a = *(const v2f*)(xrow + k);

        // B 4x16 fragments: b[0] = W[k+2*half][n0+l15], b[1] = next K row
        const float* wb = Wr + (size_t)(k + 2 * half) * 64 + l15;
        v2f b0 = { wb[ 0], wb[ 0 + 64] };
        v2f b1 = { wb[16], wb[16 + 64] };
        v2f b2 = { wb[32], wb[32 + 64] };
        v2f b3 = { wb[48], wb[48 + 64] };

        c0 = __builtin_amdgcn_wmma_f32_16x16x4_f32(false, a, false, b0, (short)0, c0, false, false);
        c1 = __builtin_amdgcn_wmma_f32_16x16x4_f32(false, a, false, b1, (short)0, c1, false, false);
        c2 = __builtin_amdgcn_wmma_f32_16x16x4_f32(false, a, false, b2, (short)0, c2, false, false);
        c3 = __builtin_amdgcn_wmma_f32_16x16x4_f32(false, a, false, b3, (short)0, c3, false, false);
    }

    // C/D layout: lane gives column n0+l15; VGPR v is row m0 + 8*half + v.
    if (r < 8) {
        float* o = h_all + (size_t)(m0 + 8 * half) * 512 + (size_t)r * 64 + l15;
#pragma unroll
        for (int v = 0; v < 8; ++v) {
            if (m0 + 8 * half + v < n_nodes) {
                o[ 0] = c0[v];
                o[16] = c1[v];
                o[32] = c2[v];
                o[48] = c3[v];
            }
            o += 512;
        }
    } else {
        const float bb0 = bias[l15];
        const float bb1 = bias[l15 + 16];
        const float bb2 = bias[l15 + 32];
        const float bb3 = bias[l15 + 48];
        float* o = agg + (size_t)(m0 + 8 * half) * 64 + l15;
#pragma unroll
        for (int v = 0; v < 8; ++v) {
            if (m0 + 8 * half + v < n_nodes) {
                o[ 0] = c0[v] + bb0;
                o[16] = c1[v] + bb1;
                o[32] = c2[v] + bb2;
                o[48] = c3[v] + bb3;
            }
            o += 64;
        }
    }
}

// ---------------------------------------------------------------------------
// Kernel 2: edge scatter. Thread per (edge, channel): gather h_all[src,etype]
// and atomically accumulate into agg[dst]. agg fits in L2 (12.8 MB / 192 MB),
// so the f32 atomic RMW resolves in-cache.
// ---------------------------------------------------------------------------
__global__ __launch_bounds__(256) void rgcn_scatter(
    const float* __restrict__ h_all,   // [N, 8, 64]
    const int* __restrict__ src,
    const int* __restrict__ dst,
    const int* __restrict__ etype,
    float* __restrict__ agg,           // [N, 64]
    int n_edges)
{
    const long long gid = (long long)blockIdx.x * blockDim.x + threadIdx.x;
    const int e = (int)(gid >> 6);
    if (e >= n_edges) return;
    const int ch = (int)(gid & 63);

    const float msg = h_all[(size_t)src[e] * 512 + (size_t)etype[e] * 64 + ch];
    atomicAdd(agg + (size_t)dst[e] * 64 + ch, msg);
}

// ---------------------------------------------------------------------------
// Kernel 3: history splice -> final output.
// ---------------------------------------------------------------------------
__global__ __launch_bounds__(256) void rgcn_splice(
    const float* __restrict__ agg,       // [N, 64]
    const float* __restrict__ hist_buf,  // [BUF, 64]
    const int* __restrict__ hist_map,    // [N]
    float* __restrict__ out,             // [N, 64]
    int n_nodes)
{
    const long long gid = (long long)blockIdx.x * blockDim.x + threadIdx.x;
    const int n = (int)(gid >> 6);
    if (n >= n_nodes) return;
    const int ch = (int)(gid & 63);

    const int hm = hist_map[n];
    const float v = (hm >= 0) ? hist_buf[(size_t)hm * 64 + ch]
                              : agg[(size_t)n * 64 + ch];
    out[(size_t)n * 64 + ch] = v;
}

// ---------------------------------------------------------------------------
// Host-side launcher. Input order (setup_inputs dict):
//   0 x [N,64] f32        1 W [8,64,64] f32    2 loop_w [64,64] f32
//   3 bias [64] f32       4 history_buffer [BUF,64] f32
//   5 src [E] i32         6 dst [E] i32        7 etypes [E] i32
//   8 history_map [N] i32
// ---------------------------------------------------------------------------
extern "C" void kernel_launch(void* const* d_in, const int* in_sizes, int n_in,
                              void* d_out, int out_size, void* d_ws, size_t ws_size,
                              hipStream_t stream) {
    const float* x      = (const float*)d_in[0];
    const float* W      = (const float*)d_in[1];
    const float* loop_w = (const float*)d_in[2];
    const float* bias   = (const float*)d_in[3];
    const float* hbuf   = (const float*)d_in[4];
    const int*   src    = (const int*)d_in[5];
    const int*   dst    = (const int*)d_in[6];
    const int*   etype  = (const int*)d_in[7];
    const int*   hmap   = (const int*)d_in[8];

    const int N = in_sizes[0] / 64;   // 50000
    const int E = in_sizes[5];        // 800000

    // Workspace layout: h_all [N*512] floats, then agg [N*64] floats (~115 MB)
    float* h_all = (float*)d_ws;
    float* agg   = h_all + (size_t)N * 512;

    // Kernel 1: WMMA transforms (relations 0..7 -> h_all, 8 -> agg init)
    const int n_tiles = (N + 15) / 16;
    const int jobs = n_tiles * 9;
    const int waves_per_block = 256 / 32;
    const int gridA = (jobs + waves_per_block - 1) / waves_per_block;
    rgcn_mm<<<gridA, 256, 0, stream>>>(x, W, loop_w, bias, h_all, agg, n_tiles, N);

    // Kernel 2: edge gather + atomic scatter-sum
    const long long sthreads = (long long)E * 64;
    const int gridB = (int)((sthreads + 255) / 256);
    rgcn_scatter<<<gridB, 256, 0, stream>>>(h_all, src, dst, etype, agg, E);

    // Kernel 3: history splice into d_out
    const long long othreads = (long long)N * 64;
    const int gridC = (int)((othreads + 255) / 256);
    rgcn_splice<<<gridC, 256, 0, stream>>>(agg, hbuf, hmap, (float*)d_out, N);
}